// MultiscaleTensorDecom_13726715478498
// MI455X (gfx1250) — compile-verified
//
#include <hip/hip_runtime.h>

#define NLEV 4
#define NF   16
#define NPTS 262144

typedef float f32x2 __attribute__((ext_vector_type(2)));
typedef float f32x4 __attribute__((ext_vector_type(4)));

// 8-byte pair load with only 4-byte alignment guarantee (x0 may be odd).
// __builtin_memcpy lets the backend emit an align-4 b64 load (verified:
// pair loads lower to single vmem ops, not 2x b32).
__device__ __forceinline__ f32x2 ld2v(const float* p) {
    f32x2 v;
    __builtin_memcpy(&v, p, sizeof(f32x2));
    return v;
}

struct Params {
    const float* pts;
    const float* pxy[NLEV];
    const float* pyz[NLEV];
    const float* pxz[NLEV];
    const float* lx[NLEV];
    const float* ly[NLEV];
    const float* lz[NLEV];
    float* out;
    int R[NLEV];
};

// Bilinear sample of 8 consecutive channels of an (NF, R, R) channel-major
// plane. Each b64 load gives a (v_col0, v_col1) pair in adjacent VGPRs, the
// exact operand shape for V_PK_MUL_F32 / V_PK_FMA_F32:
//   acc = t0 * (w00,w01) + t1 * (w10,w11);  out = acc.x + acc.y
__device__ __forceinline__ void plane8(const float* __restrict__ base,
                                       int R, int rr, int off,
                                       f32x2 wA, f32x2 wB, float out[8]) {
    f32x2 t0[8], t1[8];
    int o = off;
#pragma unroll
    for (int k = 0; k < 8; ++k) {
        t0[k] = ld2v(base + o);       // row0: (v00, v01)
        t1[k] = ld2v(base + o + R);   // row1: (v10, v11)
        o += rr;
    }
#pragma unroll
    for (int k = 0; k < 8; ++k) {
        f32x2 acc = t0[k] * wA + t1[k] * wB;
        out[k] = acc.x + acc.y;
    }
}

// Linear sample of 8 consecutive channels of an (NF, R) line.
__device__ __forceinline__ void line8(const float* __restrict__ base,
                                      int R, int off, f32x2 wl, float out[8]) {
    f32x2 t[8];
    int o = off;
#pragma unroll
    for (int k = 0; k < 8; ++k) { t[k] = ld2v(base + o); o += R; }
#pragma unroll
    for (int k = 0; k < 8; ++k) {
        f32x2 a = t[k] * wl;
        out[k] = a.x + a.y;
    }
}

__global__ __launch_bounds__(256)
void MultiscaleTensorDecom_kernel(Params P) {
    const int t  = blockIdx.x * 256 + threadIdx.x;   // 2 threads per point
    const int p  = t >> 1;
    const int c0 = (t & 1) * 8;                      // channel octet
    if (p >= NPTS) return;

    const float x1 = P.pts[p * 3 + 0] + 1.0f;
    const float y1 = P.pts[p * 3 + 1] + 1.0f;
    const float z1 = P.pts[p * 3 + 2] + 1.0f;

#pragma unroll
    for (int l = 0; l < NLEV; ++l) {
        const int   R  = P.R[l];
        const int   rr = R * R;
        const float s  = 0.5f * (float)(R - 1);
        const float fx = x1 * s;
        const float fy = y1 * s;
        const float fz = z1 * s;

        // floor (coords are non-negative) + clamp to R-2 so i1 = i0+1 always.
        // (Clamp only fires when fp rounding pushes f* to exactly R-1, where
        // w*=1.0 then selects the same texel the reference's clip does.)
        int ix = (int)fx; ix = ix > (R - 2) ? (R - 2) : ix;
        int iy = (int)fy; iy = iy > (R - 2) ? (R - 2) : iy;
        int iz = (int)fz; iz = iz > (R - 2) ? (R - 2) : iz;
        const float wx = fx - (float)ix, ux = 1.0f - wx;
        const float wy = fy - (float)iy, uy = 1.0f - wy;
        const float wz = fz - (float)iz, uz = 1.0f - wz;

        // Packed weight pairs (col-weight pairs match the b64 data pairs).
        const f32x2 wyzA = {uy * uz, wy * uz}, wyzB = {uy * wz, wy * wz};
        const f32x2 wxzA = {ux * uz, wx * uz}, wxzB = {ux * wz, wx * wz};
        const f32x2 wxyA = {ux * uy, wx * uy}, wxyB = {ux * wy, wx * wy};
        const f32x2 wlx  = {ux, wx}, wly = {uy, wy}, wlz = {uz, wz};

        float fyz[8], fxz[8], fxy[8], gx[8], gy[8], gz[8];
        // f_yz: col <- y, row <- z ; f_xz: col <- x, row <- z ; f_xy: col <- x, row <- y
        plane8(P.pyz[l], R, rr, c0 * rr + iz * R + iy, wyzA, wyzB, fyz);
        plane8(P.pxz[l], R, rr, c0 * rr + iz * R + ix, wxzA, wxzB, fxz);
        plane8(P.pxy[l], R, rr, c0 * rr + iy * R + ix, wxyA, wxyB, fxy);
        line8(P.lx[l], R, c0 * R + ix, wlx, gx);
        line8(P.ly[l], R, c0 * R + iy, wly, gy);
        line8(P.lz[l], R, c0 * R + iz, wlz, gz);

        // Streaming, write-once output -> non-temporal b128 stores (keep the
        // L2-resident planes from being evicted by 201 MB of output traffic).
        const int ob = (3 * l) * (NPTS * NF) + p * NF + c0;
        f32x4 v;

        v = (f32x4){fyz[0] * gx[0], fyz[1] * gx[1], fyz[2] * gx[2], fyz[3] * gx[3]};
        __builtin_nontemporal_store(v, (f32x4*)(P.out + ob));
        v = (f32x4){fyz[4] * gx[4], fyz[5] * gx[5], fyz[6] * gx[6], fyz[7] * gx[7]};
        __builtin_nontemporal_store(v, (f32x4*)(P.out + ob + 4));

        v = (f32x4){fxz[0] * gy[0], fxz[1] * gy[1], fxz[2] * gy[2], fxz[3] * gy[3]};
        __builtin_nontemporal_store(v, (f32x4*)(P.out + ob + (NPTS * NF)));
        v = (f32x4){fxz[4] * gy[4], fxz[5] * gy[5], fxz[6] * gy[6], fxz[7] * gy[7]};
        __builtin_nontemporal_store(v, (f32x4*)(P.out + ob + (NPTS * NF) + 4));

        v = (f32x4){fxy[0] * gz[0], fxy[1] * gz[1], fxy[2] * gz[2], fxy[3] * gz[3]};
        __builtin_nontemporal_store(v, (f32x4*)(P.out + ob + 2 * (NPTS * NF)));
        v = (f32x4){fxy[4] * gz[4], fxy[5] * gz[5], fxy[6] * gz[6], fxy[7] * gz[7]};
        __builtin_nontemporal_store(v, (f32x4*)(P.out + ob + 2 * (NPTS * NF) + 4));
    }
}

extern "C" void kernel_launch(void* const* d_in, const int* in_sizes, int n_in,
                              void* d_out, int out_size, void* d_ws, size_t ws_size,
                              hipStream_t stream) {
    (void)n_in; (void)out_size; (void)d_ws; (void)ws_size;

    Params P;
    P.pts = (const float*)d_in[0];
    for (int l = 0; l < NLEV; ++l) {
        P.pxy[l] = (const float*)d_in[1 + l];
        P.pyz[l] = (const float*)d_in[5 + l];
        P.pxz[l] = (const float*)d_in[9 + l];
        P.lx[l]  = (const float*)d_in[13 + l];
        P.ly[l]  = (const float*)d_in[17 + l];
        P.lz[l]  = (const float*)d_in[21 + l];
        P.R[l]   = in_sizes[13 + l] / NF;   // lines are (NF, R)
    }
    P.out = (float*)d_out;

    const int threads = NPTS * 2;           // 2 channel-octets per point
    dim3 block(256);
    dim3 grid(threads / 256);
    MultiscaleTensorDecom_kernel<<<grid, block, 0, stream>>>(P);
}